// BeamlineModel_44873818308717
// MI455X (gfx1250) — compile-verified
//
#include <hip/hip_runtime.h>
#include <math.h>
#include <stdint.h>

#define NTHREADS 256
#define NBLOCKS  512
#define KMAX     64

// ---- drift: only x/y matter for the output; z path is dead code ----
__device__ __forceinline__ void drift_xy(float& x, float& y,
                                         float px, float py,
                                         float inv_p, float L)
{
    float Px = px * inv_p;
    float Py = py * inv_p;
    float pxy2 = Px * Px + Py * Py;
    float inv_pl = rsqrtf(1.0f - pxy2);     // 1/Pl
    x += L * Px * inv_pl;
    y += L * Py * inv_pl;
}

__global__ void __launch_bounds__(NTHREADS)
beamline_track_reduce(const float* __restrict__ x_in,
                      const float* __restrict__ px_in,
                      const float* __restrict__ y_in,
                      const float* __restrict__ py_in,
                      const float* __restrict__ pz_in,
                      const float* __restrict__ k_set,
                      const int*   __restrict__ n_slices_ptr,
                      int n_part, int n_k,
                      double* __restrict__ partials)
{
    __shared__ float  ks[KMAX];
    __shared__ double red[NTHREADS];

    // ---- Stage k_set into LDS via the CDNA5 async global->LDS data path ----
    // (CDNA5 ISA: GLOBAL_LOAD_ASYNC_TO_LDS_B32, ASYNCcnt)
    if (threadIdx.x < (unsigned)n_k) {
        uint32_t lds_addr = (uint32_t)(uintptr_t)(&ks[threadIdx.x]);
        uint64_t gaddr    = (uint64_t)(uintptr_t)(k_set + threadIdx.x);
        asm volatile("global_load_async_to_lds_b32 %0, %1, off"
                     :: "v"(lds_addr), "v"(gaddr) : "memory");
    }
    asm volatile("s_wait_asynccnt 0" ::: "memory");
    __syncthreads();

    const int   n_slices = *n_slices_ptr;              // uniform scalar load
    const float l        = (float)(0.1 / (double)n_slices);   // L_Q / n_slices
    const float half     = 0.45f;                      // L_D / 2
    const float EPSF     = 2.2204460492503131e-16f;

    double sx = 0.0, sxx = 0.0, sy = 0.0, syy = 0.0;

    const int stride = gridDim.x * blockDim.x;
    for (int i = blockIdx.x * blockDim.x + threadIdx.x; i < n_part; i += stride) {
        float x  = x_in[i];
        float px = px_in[i];
        float y  = y_in[i];
        float py = py_in[i];
        float pz = pz_in[i];
        float rel_p = 1.0f + pz;                       // pz never changes
        float inv_p = 1.0f / rel_p;

        for (int q = 0; q < n_k; ++q) {
            float k1 = ks[q];

            drift_xy(x, y, px, py, inv_p, half);

            // ---- quad transfer matrix: constant across all slices (pz fixed) ----
            float k1n    = k1 * inv_p;
            float ak     = fabsf(k1n) + EPSF;
            float sk     = sqrtf(ak);
            float inv_sk = 1.0f / sk;
            float t      = sk * l;                     // |t| <~ 0.05 always
            float t2     = t * t;

            // Taylor series: full f32 accuracy for |t| < ~0.6
            float c  = 1.0f + t2 * (-0.5f + t2 * ((1.0f/24.0f)
                     + t2 * ((-1.0f/720.0f) + t2 * (1.0f/40320.0f))));
            float ch = 1.0f + t2 * ( 0.5f + t2 * ((1.0f/24.0f)
                     + t2 * (( 1.0f/720.0f) + t2 * (1.0f/40320.0f))));
            float s  = t * (1.0f + t2 * ((-1.0f/6.0f)
                     + t2 * ((1.0f/120.0f) + t2 * (-1.0f/5040.0f))));
            float sh = t * (1.0f + t2 * (( 1.0f/6.0f)
                     + t2 * ((1.0f/120.0f) + t2 * ( 1.0f/5040.0f))));

            // x-plane: quad_mat2_calc(-k1n): focus iff (-k1n) <= 0
            bool  fx   = (-k1n) <= 0.0f;
            float cx   = fx ? c : ch;
            float sxm  = (fx ? s : sh) * inv_sk;
            float tx12 = sxm * inv_p;
            float tx21 = (-k1n) * sxm * rel_p;

            // y-plane: quad_mat2_calc(+k1n): focus iff k1n <= 0
            bool  fy   = k1n <= 0.0f;
            float cy   = fy ? c : ch;
            float sym  = (fy ? s : sh) * inv_sk;
            float ty12 = sym * inv_p;
            float ty21 = k1n * sym * rel_p;

            for (int sl = 0; sl < n_slices; ++sl) {
                float nx  = cx * x  + tx12 * px;
                float npx = tx21 * x + cx * px;
                float ny  = cy * y  + ty12 * py;
                float npy = ty21 * y + cy * py;
                x = nx; px = npx; y = ny; py = npy;
            }

            drift_xy(x, y, px, py, inv_p, half);
        }

        sx  += (double)x;
        sxx += (double)x * (double)x;
        sy  += (double)y;
        syy += (double)y * (double)y;
    }

    // ---- deterministic block reduction (f64) ----
    double vals[4] = {sx, sxx, sy, syy};
    for (int q = 0; q < 4; ++q) {
        red[threadIdx.x] = vals[q];
        __syncthreads();
        for (int off = NTHREADS / 2; off > 0; off >>= 1) {
            if ((int)threadIdx.x < off) red[threadIdx.x] += red[threadIdx.x + off];
            __syncthreads();
        }
        if (threadIdx.x == 0) partials[4 * blockIdx.x + q] = red[0];
        __syncthreads();
    }
}

__global__ void __launch_bounds__(NTHREADS)
beamline_finalize(const double* __restrict__ partials, int nblk,
                  int n_part, float* __restrict__ out)
{
    __shared__ double red[NTHREADS];
    __shared__ double tot[4];

    double s[4] = {0.0, 0.0, 0.0, 0.0};
    for (int b = threadIdx.x; b < nblk; b += NTHREADS) {
        s[0] += partials[4 * b + 0];
        s[1] += partials[4 * b + 1];
        s[2] += partials[4 * b + 2];
        s[3] += partials[4 * b + 3];
    }
    for (int q = 0; q < 4; ++q) {
        red[threadIdx.x] = s[q];
        __syncthreads();
        for (int off = NTHREADS / 2; off > 0; off >>= 1) {
            if ((int)threadIdx.x < off) red[threadIdx.x] += red[threadIdx.x + off];
            __syncthreads();
        }
        if (threadIdx.x == 0) tot[q] = red[0];
        __syncthreads();
    }
    if (threadIdx.x == 0) {
        double N  = (double)n_part;
        double vx = (tot[1] - tot[0] * tot[0] / N) / (N - 1.0);
        double vy = (tot[3] - tot[2] * tot[2] / N) / (N - 1.0);
        double dx = sqrt(vx) - 0.005;
        double dy = sqrt(vy) - 0.005;
        out[0] = (float)sqrt(dx * dx + dy * dy);
    }
}

extern "C" void kernel_launch(void* const* d_in, const int* in_sizes, int n_in,
                              void* d_out, int out_size, void* d_ws, size_t ws_size,
                              hipStream_t stream)
{
    const float* x     = (const float*)d_in[0];
    const float* px    = (const float*)d_in[1];
    const float* y     = (const float*)d_in[2];
    const float* py    = (const float*)d_in[3];
    // d_in[4] = z : dead — z never feeds back into x/y and output uses only std(x), std(y)
    const float* pz    = (const float*)d_in[5];
    const float* k_set = (const float*)d_in[6];
    const int* n_slices = (const int*)d_in[7];

    int n_part = in_sizes[0];
    int n_k    = in_sizes[6];
    if (n_k > KMAX) n_k = KMAX;

    double* partials = (double*)d_ws;   // 512 blocks * 4 doubles = 16 KB

    beamline_track_reduce<<<NBLOCKS, NTHREADS, 0, stream>>>(
        x, px, y, py, pz, k_set, n_slices, n_part, n_k, partials);
    beamline_finalize<<<1, NTHREADS, 0, stream>>>(
        partials, NBLOCKS, n_part, (float*)d_out);
}